// TransformationNet_45707041964760
// MI455X (gfx1250) — compile-verified
//
#include <hip/hip_runtime.h>

#define INPUT_DIM 512
#define DIM 512
#define N_ACTIONS 128
#define BATCH 2048
#define SEQ 128
#define GN_EPS 1e-5f
#define LDSROW 516 // 512 + 4-float pad: row stride 2064B -> conflict-free banks

typedef __attribute__((ext_vector_type(2))) float v2f;
typedef __attribute__((ext_vector_type(4))) float v4f;
typedef __attribute__((ext_vector_type(8))) float v8f;

// ---------------------------------------------------------------------------
// Stage 1: masked mean over SEQ + GroupNorm(1,1) over features.
// grid (BATCH, 2): z=0 precondition -> p_avg, z=1 effect -> e_avg.
// ---------------------------------------------------------------------------
__global__ __launch_bounds__(256) void reduce_gn_kernel(
    const float* __restrict__ pre, const float* __restrict__ eff,
    const float* __restrict__ gamma, const float* __restrict__ beta,
    float* __restrict__ p_avg, float* __restrict__ e_avg) {
  const int b = blockIdx.x;
  const int z = blockIdx.y;
  const float* x = (z ? eff : pre) + (size_t)b * (SEQ * INPUT_DIM);
  float* dst = (z ? e_avg : p_avg) + (size_t)b * INPUT_DIM;
  const int t = threadIdx.x;
  const int f = 2 * t;

  float sx = 0.f, sy = 0.f, cx = 0.f, cy = 0.f;
  for (int s = 0; s < SEQ; ++s) {
    v2f v = *(const v2f*)(x + s * INPUT_DIM + f);
    sx += v.x;
    sy += v.y;
    cx += (v.x != 0.0f) ? 1.0f : 0.0f;
    cy += (v.y != 0.0f) ? 1.0f : 0.0f;
  }
  const float ax = sx / cx;
  const float ay = sy / cy;

  __shared__ float rs[256];
  __shared__ float rq[256];
  rs[t] = ax + ay;
  rq[t] = ax * ax + ay * ay;
  __syncthreads();
  for (int off = 128; off > 0; off >>= 1) {
    if (t < off) {
      rs[t] += rs[t + off];
      rq[t] += rq[t + off];
    }
    __syncthreads();
  }
  const float m = rs[0] * (1.0f / INPUT_DIM);
  const float var = rq[0] * (1.0f / INPUT_DIM) - m * m;
  const float r = rsqrtf(var + GN_EPS);
  const float g = gamma[0], be = beta[0];
  v2f o;
  o.x = g * (ax - m) * r + be;
  o.y = g * (ay - m) * r + be;
  *(v2f*)(dst + f) = o;
}

// ---------------------------------------------------------------------------
// Binning: group samples by action so the matvecs become GEMMs.
// ---------------------------------------------------------------------------
__global__ void zero_counts_kernel(int* counts) { counts[threadIdx.x] = 0; }

__global__ void bin_kernel(const int* __restrict__ action,
                           int* __restrict__ counts, int* __restrict__ lists) {
  const int b = blockIdx.x * blockDim.x + threadIdx.x;
  if (b < BATCH) {
    const int a = action[b];
    const int pos = atomicAdd(&counts[a], 1);
    lists[a * BATCH + pos] = b;
  }
}

// ---------------------------------------------------------------------------
// WMMA core: this wave's K-chunk of a 16x16 f32 tile, K step 4.
// A lane l: row M=l&15, floats (k+2*(l>>4), +1). B symmetric with N=l&15.
// ---------------------------------------------------------------------------
__device__ __forceinline__ v8f wmma_kchunk(const float* As, const float* Bs,
                                           int lane, int k0, int k1) {
  v8f c = {0.f, 0.f, 0.f, 0.f, 0.f, 0.f, 0.f, 0.f};
  const float* ar = As + (lane & 15) * LDSROW + 2 * (lane >> 4);
  const float* br = Bs + (lane & 15) * LDSROW + 2 * (lane >> 4);
  for (int k = k0; k < k1; k += 4) {
    v2f a = *(const v2f*)(ar + k);
    v2f b = *(const v2f*)(br + k);
    c = __builtin_amdgcn_wmma_f32_16x16x4_f32(false, a, false, b, (short)0, c,
                                              false, false);
  }
  return c;
}

// Stage 16 rows x 512 f32 into LDS; thread t owns float4 column-chunk t.
__device__ __forceinline__ void load_tile_rows(float* dstLds,
                                               const float* __restrict__ src,
                                               int row_stride, int t) {
#pragma unroll
  for (int i = 0; i < 16; ++i) {
    v4f v = *(const v4f*)(src + (size_t)i * row_stride + t * 4);
    *(v4f*)(dstLds + i * LDSROW + t * 4) = v;
  }
}

// ---------------------------------------------------------------------------
// Stage 2: embeddings. out[n][m] = dot(Wrow_m, avg_n) + bias[m].
// grid (DIM/16, BATCH/16, 2). 128 threads = 4 waves splitting K=512.
// ---------------------------------------------------------------------------
__global__ __launch_bounds__(128) void embed_gemm_kernel(
    const float* __restrict__ p_avg, const float* __restrict__ e_avg,
    const float* __restrict__ pw, const float* __restrict__ pb,
    const float* __restrict__ ew, const float* __restrict__ eb,
    float* __restrict__ p_embed, float* __restrict__ e_embed) {
  __shared__ float As[16 * LDSROW];
  __shared__ float Bs[16 * LDSROW];
  __shared__ float Cred[4 * 32 * 8];

  const int t = threadIdx.x;
  const int lane = t & 31;
  const int wave = t >> 5;
  const int m0 = blockIdx.x * 16; // output-dim tile
  const int n0 = blockIdx.y * 16; // batch tile
  const int z = blockIdx.z;

  const float* Wm = (z ? ew : pw) + (size_t)m0 * INPUT_DIM;
  const float* Bv = (z ? e_avg : p_avg);
  const float* bias = (z ? eb : pb);
  float* out = (z ? e_embed : p_embed);

  load_tile_rows(As, Wm, INPUT_DIM, t);
#pragma unroll
  for (int i = 0; i < 16; ++i) {
    v4f v = *(const v4f*)(Bv + (size_t)(n0 + i) * INPUT_DIM + t * 4);
    *(v4f*)(Bs + i * LDSROW + t * 4) = v;
  }
  __syncthreads();

  v8f c = wmma_kchunk(As, Bs, lane, wave * 128, wave * 128 + 128);

  float* cr = Cred + (wave * 32 + lane) * 8;
#pragma unroll
  for (int r = 0; r < 8; ++r) cr[r] = c[r];
  __syncthreads();

  if (wave == 0) {
#pragma unroll
    for (int r = 0; r < 8; ++r)
      c[r] = Cred[lane * 8 + r] + Cred[(32 + lane) * 8 + r] +
             Cred[(64 + lane) * 8 + r] + Cred[(96 + lane) * 8 + r];
    const int sample = n0 + (lane & 15);
    const int d0 = m0 + 8 * (lane >> 4);
    const float* bp = bias + d0;
#pragma unroll
    for (int r = 0; r < 8; ++r) c[r] += bp[r];
    float* o = out + (size_t)sample * DIM + d0;
    v4f lo = {c[0], c[1], c[2], c[3]};
    v4f hi = {c[4], c[5], c[6], c[7]};
    *(v4f*)(o) = lo;
    *(v4f*)(o + 4) = hi;
  }
}

// ---------------------------------------------------------------------------
// Stage 3: grouped GEMM: for action a, out[s][m] = dot(W[a]row_m, p_embed_s).
// grid (DIM/16, N_ACTIONS). W tile loaded once, reused across sample tiles.
// ---------------------------------------------------------------------------
__global__ __launch_bounds__(128) void action_gemm_kernel(
    const float* __restrict__ W, const float* __restrict__ p_embed,
    const int* __restrict__ counts, const int* __restrict__ lists,
    float* __restrict__ out) {
  __shared__ float As[16 * LDSROW];
  __shared__ float Bs[16 * LDSROW];
  __shared__ float Cred[4 * 32 * 8];
  __shared__ int samp[16];

  const int t = threadIdx.x;
  const int lane = t & 31;
  const int wave = t >> 5;
  const int m0 = blockIdx.x * 16;
  const int a = blockIdx.y;

  const int cnt = counts[a];
  if (cnt == 0) return;
  const int ntiles = (cnt + 15) >> 4;

  const float* Wa = W + (size_t)a * DIM * DIM + (size_t)m0 * DIM;
  load_tile_rows(As, Wa, DIM, t);

  for (int nt = 0; nt < ntiles; ++nt) {
    __syncthreads(); // covers As load (nt=0) and prior-iter LDS reads
    if (t < 16) {
      const int idx = nt * 16 + t;
      samp[t] = (idx < cnt) ? lists[a * BATCH + idx] : -1;
    }
    __syncthreads();
#pragma unroll
    for (int i = 0; i < 16; ++i) {
      const int s = samp[i];
      v4f v = {0.f, 0.f, 0.f, 0.f};
      if (s >= 0) v = *(const v4f*)(p_embed + (size_t)s * DIM + t * 4);
      *(v4f*)(Bs + i * LDSROW + t * 4) = v;
    }
    __syncthreads();

    v8f c = wmma_kchunk(As, Bs, lane, wave * 128, wave * 128 + 128);
    float* cr = Cred + (wave * 32 + lane) * 8;
#pragma unroll
    for (int r = 0; r < 8; ++r) cr[r] = c[r];
    __syncthreads();

    if (wave == 0) {
#pragma unroll
      for (int r = 0; r < 8; ++r)
        c[r] = Cred[lane * 8 + r] + Cred[(32 + lane) * 8 + r] +
               Cred[(64 + lane) * 8 + r] + Cred[(96 + lane) * 8 + r];
      const int s = samp[lane & 15];
      if (s >= 0) {
        float* o = out + (size_t)s * DIM + m0 + 8 * (lane >> 4);
        v4f lo = {c[0], c[1], c[2], c[3]};
        v4f hi = {c[4], c[5], c[6], c[7]};
        *(v4f*)(o) = lo;
        *(v4f*)(o + 4) = hi;
      }
    }
  }
}

// ---------------------------------------------------------------------------
extern "C" void kernel_launch(void* const* d_in, const int* in_sizes, int n_in,
                              void* d_out, int out_size, void* d_ws,
                              size_t ws_size, hipStream_t stream) {
  (void)in_sizes; (void)n_in; (void)out_size; (void)ws_size;
  const float* pre = (const float*)d_in[0];
  const float* eff = (const float*)d_in[1];
  const int* action = (const int*)d_in[2];
  const float* W = (const float*)d_in[3];
  const float* pw = (const float*)d_in[4];
  const float* pb = (const float*)d_in[5];
  const float* ew = (const float*)d_in[6];
  const float* eb = (const float*)d_in[7];
  const float* gamma = (const float*)d_in[8];
  const float* beta = (const float*)d_in[9];

  float* out_pt = (float*)d_out;                 // p_transformed [BATCH,DIM]
  float* out_ee = out_pt + (size_t)BATCH * DIM;  // e_embed       [BATCH,DIM]

  float* p_avg = (float*)d_ws;
  float* e_avg = p_avg + (size_t)BATCH * INPUT_DIM;
  float* p_embed = e_avg + (size_t)BATCH * INPUT_DIM;
  int* counts = (int*)(p_embed + (size_t)BATCH * DIM);
  int* lists = counts + 256; // 128 counters, padded

  reduce_gn_kernel<<<dim3(BATCH, 2), 256, 0, stream>>>(pre, eff, gamma, beta,
                                                       p_avg, e_avg);
  zero_counts_kernel<<<1, N_ACTIONS, 0, stream>>>(counts);
  bin_kernel<<<BATCH / 256, 256, 0, stream>>>(action, counts, lists);
  embed_gemm_kernel<<<dim3(DIM / 16, BATCH / 16, 2), 128, 0, stream>>>(
      p_avg, e_avg, pw, pb, ew, eb, p_embed, out_ee);
  action_gemm_kernel<<<dim3(DIM / 16, N_ACTIONS), 128, 0, stream>>>(
      W, p_embed, counts, lists, out_pt);
}